// RPN_76862734729533
// MI455X (gfx1250) — compile-verified
//
#include <hip/hip_runtime.h>
#include <math.h>
#include <stdint.h>

// ---------------- CDNA5 vector types ----------------
typedef float v2f __attribute__((ext_vector_type(2)));
typedef float v4f __attribute__((ext_vector_type(4)));
typedef float v8f __attribute__((ext_vector_type(8)));

#define ICH 512
#define OCH 512
#define FH 100
#define FW 100
#define PH 102             // padded height
#define PW 102             // padded width
#define NPP (PH * PW)      // padded positions
#define NP 10000           // 100*100 positions
#define NANCH 9
#define NPROP 90000        // NP * NANCH
#define NSORT 131072       // next pow2 >= NPROP
#define TOPK 2000
#define NWORDS 32          // ceil(2000/64)
#define BBOX_CLIP 4.135166556742356f
#define IMSZ 1600.0f
#define MIN_SIZE 16.0f
#define IOU_T 0.7f

struct AnchorBases { float b[36]; };

// =====================================================================
// One-time data movement (≈50 MB total; ~2us at 23.3 TB/s HBM):
//  - Xt[pp][ic]: zero-padded (102x102), channel-minor features
//  - Wp[oc][tap][ic]: tap-major, channel-minor weights
// Both make each lane's WMMA K-pair a contiguous 8-byte load and remove
// every bounds check from the conv inner loop (EXEC stays all-ones).
// =====================================================================
__global__ void zero_buf(float* __restrict__ buf, int n)
{
  const int i = blockIdx.x * blockDim.x + threadIdx.x;
  if (i < n) buf[i] = 0.0f;
}

__global__ __launch_bounds__(256) void transpose_feat(
    const float* __restrict__ feat,   // [512][100][100]
    float* __restrict__ Xt)           // [102*102][512], border pre-zeroed
{
  __shared__ float tile[64][65];
  const int p0  = blockIdx.x * 64;
  const int icb = blockIdx.y * 64;
  const int tid = threadIdx.x;
#pragma unroll
  for (int pass = 0; pass < 16; ++pass) {
    const int idx = pass * 256 + tid;
    const int r = idx >> 6;        // ic sub-index
    const int c = idx & 63;        // pos sub-index
    const int p = p0 + c;
    tile[r][c] = (p < NP) ? feat[(icb + r) * NP + p] : 0.0f;
  }
  __syncthreads();
#pragma unroll
  for (int pass = 0; pass < 16; ++pass) {
    const int idx = pass * 256 + tid;
    const int c = idx >> 6;        // pos sub-index
    const int r = idx & 63;        // ic sub-index (fastest -> coalesced writes)
    const int p = p0 + c;
    if (p < NP) {
      const int y = p / FW, x = p % FW;
      const int pp = (y + 1) * PW + (x + 1);
      Xt[(size_t)pp * ICH + icb + r] = tile[r][c];
    }
  }
}

__global__ void pack_weights(const float* __restrict__ Wsrc,  // [oc][ic][tap]
                             float* __restrict__ Wp)          // [oc][tap][ic]
{
  const int t = blockIdx.x * blockDim.x + threadIdx.x;
  if (t >= OCH * ICH * 9) return;
  const int oc  = t / (ICH * 9);
  const int rem = t - oc * (ICH * 9);
  const int tap = rem / ICH;
  const int ic  = rem - tap * ICH;
  Wp[t] = Wsrc[(oc * ICH + ic) * 9 + tap];
}

// =====================================================================
// 3x3 conv as GEMM: M=512, N=10000, K=4608 (K ordered tap-major).
// One wave = 16(M) x 64(N) tile via V_WMMA_F32_16X16X4_F32; 8 waves/block
// share the same 64 positions (WGP$/L2 reuse; whole 30MB working set is
// L2-resident on MI455X's 192MB L2).
// A layout (16x4 f32): lanes0-15 M=lane, K={0,1}; lanes16-31 K={2,3}.
// B layout (4x16 f32): lanes0-15 N=lane, K={0,1}; lanes16-31 K={2,3}.
// D layout (16x16 f32): VGPR r -> M = r + 8*(lane>=16), N = lane&15.
// Inner loop: 5 x global_load_b64 + 4 x v_wmma, branch-free.
// =====================================================================
__global__ __launch_bounds__(256) void conv3x3_wmma(
    const float* __restrict__ Xt,    // [102*102][512]
    const float* __restrict__ Wp,    // [512][9][512]
    const float* __restrict__ bias,  // [512]
    float* __restrict__ Yt)          // [10000][512]
{
  const int lane = threadIdx.x & 31;
  const int wave = threadIdx.x >> 5;
  const int oc0  = (blockIdx.x * 8 + wave) * 16;   // 0..496
  const int n0   = blockIdx.y * 64;
  const int nl   = lane & 15;
  const int kh   = lane >> 4;

  v8f acc[4] = {};
  const float* wrow = Wp + (size_t)(oc0 + nl) * (ICH * 9) + 2 * kh;

  int ppj[4];
#pragma unroll
  for (int j = 0; j < 4; ++j) {
    const int p = n0 + j * 16 + nl;          // may overhang to 10047: y<=100,
    const int y = p / FW, x = p % FW;        // stays inside padded buffer
    ppj[j] = (y + 1) * PW + (x + 1);
  }

  for (int tap = 0; tap < 9; ++tap) {
    const int dy = tap / 3 - 1;
    const int dx = tap % 3 - 1;
    const float* wtap = wrow + tap * ICH;
    const float* xb0 = Xt + (size_t)(ppj[0] + dy * PW + dx) * ICH + 2 * kh;
    const float* xb1 = Xt + (size_t)(ppj[1] + dy * PW + dx) * ICH + 2 * kh;
    const float* xb2 = Xt + (size_t)(ppj[2] + dy * PW + dx) * ICH + 2 * kh;
    const float* xb3 = Xt + (size_t)(ppj[3] + dy * PW + dx) * ICH + 2 * kh;
#pragma unroll 2
    for (int ic0 = 0; ic0 < ICH; ic0 += 4) {
      const v2f a  = *(const v2f*)(wtap + ic0);
      const v2f b0 = *(const v2f*)(xb0 + ic0);
      const v2f b1 = *(const v2f*)(xb1 + ic0);
      const v2f b2 = *(const v2f*)(xb2 + ic0);
      const v2f b3 = *(const v2f*)(xb3 + ic0);
      acc[0] = __builtin_amdgcn_wmma_f32_16x16x4_f32(false, a, false, b0, (short)0, acc[0], false, false);
      acc[1] = __builtin_amdgcn_wmma_f32_16x16x4_f32(false, a, false, b1, (short)0, acc[1], false, false);
      acc[2] = __builtin_amdgcn_wmma_f32_16x16x4_f32(false, a, false, b2, (short)0, acc[2], false, false);
      acc[3] = __builtin_amdgcn_wmma_f32_16x16x4_f32(false, a, false, b3, (short)0, acc[3], false, false);
    }
  }

  float bl[8];
#pragma unroll
  for (int r = 0; r < 8; ++r) bl[r] = bias[oc0 + 8 * kh + r];

#pragma unroll
  for (int j = 0; j < 4; ++j) {
    const int p = n0 + j * 16 + nl;
    if (p < NP) {
      float* yp = Yt + (size_t)p * OCH + oc0 + 8 * kh;
      v4f lo = {acc[j][0] + bl[0], acc[j][1] + bl[1], acc[j][2] + bl[2], acc[j][3] + bl[3]};
      v4f hi = {acc[j][4] + bl[4], acc[j][5] + bl[5], acc[j][6] + bl[6], acc[j][7] + bl[7]};
      *(v4f*)(yp)     = lo;
      *(v4f*)(yp + 4) = hi;
    }
  }
}

// =====================================================================
// 1x1 conv heads on position-major conv_out: contiguous b64 K-pair loads,
// M padded to 16 with a 0.0 multiplier (no branches, EXEC all-ones).
// Writes NHWC-flat layout the reference's transpose/reshape produces.
// =====================================================================
__global__ __launch_bounds__(32) void conv1x1_wmma(
    const float* __restrict__ Xt,   // [10000][512]
    const float* __restrict__ Wt,   // [C][512]
    const float* __restrict__ bias, // [C]
    float* __restrict__ Yout,       // [10000][C]
    int C, int applySigmoid)
{
  const int lane = threadIdx.x & 31;
  const int nl = lane & 15;
  const int kh = lane >> 4;
  const int m0 = blockIdx.x * 16;
  const int n0 = blockIdx.y * 64;
  const int m  = m0 + nl;
  const float msc = (m < C) ? 1.0f : 0.0f;
  const float* wrow = Wt + (size_t)(m < C ? m : 0) * ICH + 2 * kh;

  const float* xb[4];
#pragma unroll
  for (int j = 0; j < 4; ++j) {
    int p = n0 + j * 16 + nl;
    if (p >= NP) p = NP - 1;                 // safe address; stores are masked
    xb[j] = Xt + (size_t)p * ICH + 2 * kh;
  }

  v8f acc[4] = {};
#pragma unroll 2
  for (int k0 = 0; k0 < ICH; k0 += 4) {
    v2f a = *(const v2f*)(wrow + k0);
    a.x *= msc;
    a.y *= msc;
#pragma unroll
    for (int j = 0; j < 4; ++j) {
      const v2f b = *(const v2f*)(xb[j] + k0);
      acc[j] = __builtin_amdgcn_wmma_f32_16x16x4_f32(
          false, a, false, b, (short)0, acc[j], false, false);
    }
  }

#pragma unroll
  for (int j = 0; j < 4; ++j) {
    const int p = n0 + j * 16 + nl;
#pragma unroll
    for (int r = 0; r < 8; ++r) {
      const int mm = m0 + r + 8 * kh;
      if (mm < C && p < NP) {
        float v = acc[j][r] + bias[mm];
        if (applySigmoid) v = 1.0f / (1.0f + expf(-v));
        Yout[(size_t)p * C + mm] = v;
      }
    }
  }
}

// ---------------- exact top-k via bitonic sort of packed keys ----------------
// key = monotonic(score) << 32 | ~index   -> descending sort gives
// score-desc, index-asc (matches jax.lax.top_k tie-breaking).
__global__ void init_keys(const float* __restrict__ scores,
                          unsigned long long* __restrict__ keys)
{
  const int i = blockIdx.x * blockDim.x + threadIdx.x;
  if (i >= NSORT) return;
  unsigned long long key = 0ull;
  if (i < NPROP) {
    unsigned u = __float_as_uint(scores[i]);
    u = (u & 0x80000000u) ? ~u : (u | 0x80000000u);
    key = ((unsigned long long)u << 32) | (unsigned)(~(unsigned)i);
  }
  keys[i] = key;
}

__global__ void bitonic_step(unsigned long long* __restrict__ d, int j, int k)
{
  const int i = blockIdx.x * blockDim.x + threadIdx.x;
  const int ixj = i ^ j;
  if (ixj <= i || i >= NSORT) return;
  const unsigned long long a = d[i];
  const unsigned long long b = d[ixj];
  const bool descBlock = ((i & k) == 0);
  const bool doSwap = descBlock ? (a < b) : (a > b);
  if (doSwap) { d[i] = b; d[ixj] = a; }
}

// ---------------- decode gathered top-2000, clip, min-size filter ------------
__global__ void decode_topk(const unsigned long long* __restrict__ keys,
                            const float* __restrict__ deltas,  // [NP][36]
                            AnchorBases ab,
                            float* __restrict__ boxes,
                            float* __restrict__ sv,
                            int* __restrict__ valid)
{
  const int i = blockIdx.x * blockDim.x + threadIdx.x;
  if (i >= TOPK) return;
  const unsigned long long key = keys[i];
  const unsigned idx = ~(unsigned)(key & 0xFFFFFFFFull);
  const unsigned um  = (unsigned)(key >> 32);
  const unsigned fb  = (um & 0x80000000u) ? (um & 0x7FFFFFFFu) : ~um;
  const float score  = __uint_as_float(fb);
  if (idx >= NPROP) {
    boxes[i * 4 + 0] = boxes[i * 4 + 1] = boxes[i * 4 + 2] = boxes[i * 4 + 3] = 0.0f;
    sv[i] = -INFINITY; valid[i] = 0; return;
  }
  const int p = idx / 9;
  const int a = idx - p * 9;
  const int px = p % FW, py = p / FW;
  const float sx = px * 16.0f, sy = py * 16.0f;
  const float ax1 = sx + ab.b[a * 4 + 0];
  const float ay1 = sy + ab.b[a * 4 + 1];
  const float ax2 = sx + ab.b[a * 4 + 2];
  const float ay2 = sy + ab.b[a * 4 + 3];
  const float wa = ax2 - ax1, ha = ay2 - ay1;
  const float cxa = ax1 + 0.5f * wa, cya = ay1 + 0.5f * ha;
  const float* d = deltas + (size_t)idx * 4;   // (p*9+a)*4 == p*36 + a*4
  const float dxv = d[0], dyv = d[1];
  const float dwv = fminf(d[2], BBOX_CLIP);
  const float dhv = fminf(d[3], BBOX_CLIP);
  const float cx = dxv * wa + cxa, cy = dyv * ha + cya;
  const float w = expf(dwv) * wa, h = expf(dhv) * ha;
  float x1 = cx - 0.5f * w, y1 = cy - 0.5f * h;
  float x2 = cx + 0.5f * w, y2 = cy + 0.5f * h;
  x1 = fminf(fmaxf(x1, 0.0f), IMSZ);
  y1 = fminf(fmaxf(y1, 0.0f), IMSZ);
  x2 = fminf(fmaxf(x2, 0.0f), IMSZ);
  y2 = fminf(fmaxf(y2, 0.0f), IMSZ);
  const bool v = ((x2 - x1) >= MIN_SIZE) && ((y2 - y1) >= MIN_SIZE);
  boxes[i * 4 + 0] = x1; boxes[i * 4 + 1] = y1;
  boxes[i * 4 + 2] = x2; boxes[i * 4 + 3] = y2;
  sv[i] = v ? score : -INFINITY;
  valid[i] = v ? 1 : 0;
}

// ------- stable valid-first partition (== stable argsort(-s) on sorted s) ----
__global__ __launch_bounds__(256) void reorder_valid_first(
    const float* __restrict__ bi, const float* __restrict__ si, const int* __restrict__ vi,
    float* __restrict__ bo, float* __restrict__ so, int* __restrict__ vo)
{
  __shared__ int vflag[TOPK];
  __shared__ int pos[TOPK];
  for (int i = threadIdx.x; i < TOPK; i += 256) vflag[i] = vi[i];
  __syncthreads();
  if (threadIdx.x == 0) {
    int c = 0;
    for (int i = 0; i < TOPK; ++i) if (vflag[i])  pos[i] = c++;
    for (int i = 0; i < TOPK; ++i) if (!vflag[i]) pos[i] = c++;
  }
  __syncthreads();
  for (int i = threadIdx.x; i < TOPK; i += 256) {
    const int p = pos[i];
    bo[p * 4 + 0] = bi[i * 4 + 0];
    bo[p * 4 + 1] = bi[i * 4 + 1];
    bo[p * 4 + 2] = bi[i * 4 + 2];
    bo[p * 4 + 3] = bi[i * 4 + 3];
    so[p] = si[i];
    vo[p] = vflag[i];
  }
}

// ---------------- NMS: suppression bit-matrix then sequential reduce ---------
__global__ void nms_bits(const float* __restrict__ boxes,
                         unsigned long long* __restrict__ mat)
{
  const int t = blockIdx.x * blockDim.x + threadIdx.x;
  if (t >= TOPK * NWORDS) return;
  const int i = t >> 5;
  const int w = t & 31;
  const float x1 = boxes[i * 4 + 0], y1 = boxes[i * 4 + 1];
  const float x2 = boxes[i * 4 + 2], y2 = boxes[i * 4 + 3];
  const float ai = (x2 - x1) * (y2 - y1);
  unsigned long long bits = 0ull;
  const int jbase = w * 64;
  for (int b = 0; b < 64; ++b) {
    const int j = jbase + b;
    if (j < TOPK && j > i) {
      const float u1 = boxes[j * 4 + 0], v1 = boxes[j * 4 + 1];
      const float u2 = boxes[j * 4 + 2], v2 = boxes[j * 4 + 3];
      const float aj = (u2 - u1) * (v2 - v1);
      float ix = fminf(x2, u2) - fmaxf(x1, u1);
      float iy = fminf(y2, v2) - fmaxf(y1, v1);
      ix = fmaxf(ix, 0.0f); iy = fmaxf(iy, 0.0f);
      const float inter = ix * iy;
      const float iou = inter / (ai + aj - inter + 1e-9f);
      if (iou > IOU_T) bits |= (1ull << b);
    }
  }
  mat[i * NWORDS + w] = bits;
}

__global__ __launch_bounds__(32) void nms_reduce(
    const unsigned long long* __restrict__ mat,
    const int* __restrict__ valid, int* __restrict__ keep)
{
  __shared__ unsigned long long remv[NWORDS];
  const int lane = threadIdx.x;
  remv[lane] = 0ull;
  __syncthreads();
  for (int i = 0; i < TOPK; ++i) {
    const unsigned long long wv = remv[i >> 6];
    const bool rem = (wv >> (i & 63)) & 1ull;
    const int k = (valid[i] != 0) && !rem;
    if (lane == 0) keep[i] = k;
    __syncthreads();
    if (k) remv[lane] |= mat[i * NWORDS + lane];
    __syncthreads();
  }
}

__global__ void finalize(const float* __restrict__ boxes,
                         const float* __restrict__ sv,
                         const int* __restrict__ keep,
                         float* __restrict__ out)
{
  const int t = blockIdx.x * blockDim.x + threadIdx.x;
  if (t >= TOPK * 5) return;
  const int i = t / 5;
  const int c = t - i * 5;
  float v = 0.0f;
  if (keep[i]) v = (c < 4) ? boxes[i * 4 + c] : sv[i];
  out[t] = v;
}

// =====================================================================
extern "C" void kernel_launch(void* const* d_in, const int* in_sizes, int n_in,
                              void* d_out, int out_size, void* d_ws, size_t ws_size,
                              hipStream_t stream)
{
  (void)in_sizes; (void)n_in; (void)out_size; (void)ws_size;
  const float* feat  = (const float*)d_in[1];
  const float* rpn_w = (const float*)d_in[2];
  const float* rpn_b = (const float*)d_in[3];
  const float* reg_w = (const float*)d_in[4];
  const float* reg_b = (const float*)d_in[5];
  const float* cls_w = (const float*)d_in[6];
  const float* cls_b = (const float*)d_in[7];
  float* out = (float*)d_out;

  // workspace carve-up (~57 MB total)
  char* ws = (char*)d_ws;
  size_t off = 0;
  auto take = [&](size_t bytes) -> void* {
    void* p = ws + off;
    off = (off + bytes + 255) & ~(size_t)255;
    return p;
  };
  float* Xt     = (float*)take((size_t)NPP * ICH * 4);       // 21.3 MB
  float* Wpack  = (float*)take((size_t)OCH * ICH * 9 * 4);   // 9.4 MB
  float* convYt = (float*)take((size_t)NP * OCH * 4);        // 20.5 MB
  float* deltas = (float*)take((size_t)NP * 36 * 4);
  float* scores = (float*)take((size_t)NPROP * 4);
  unsigned long long* keys = (unsigned long long*)take((size_t)NSORT * 8);
  float* boxesA = (float*)take((size_t)TOPK * 4 * 4);
  float* sA     = (float*)take((size_t)TOPK * 4);
  int*   vA     = (int*)take((size_t)TOPK * 4);
  float* boxesB = (float*)take((size_t)TOPK * 4 * 4);
  float* sB     = (float*)take((size_t)TOPK * 4);
  int*   vB     = (int*)take((size_t)TOPK * 4);
  unsigned long long* mat = (unsigned long long*)take((size_t)TOPK * NWORDS * 8);
  int*   keep   = (int*)take((size_t)TOPK * 4);

  // anchor bases on host (rintf == round-half-even == jnp.round)
  AnchorBases ab;
  const float scales[3] = {128.0f, 256.0f, 512.0f};
  const float ratios[3] = {0.5f, 1.0f, 2.0f};
  int a = 0;
  for (int r = 0; r < 3; ++r) {
    const float hr = sqrtf(ratios[r]);
    const float wr = 1.0f / hr;
    for (int s = 0; s < 3; ++s) {
      const float wsz = wr * scales[s];
      const float hsz = hr * scales[s];
      ab.b[a * 4 + 0] = rintf(-wsz * 0.5f);
      ab.b[a * 4 + 1] = rintf(-hsz * 0.5f);
      ab.b[a * 4 + 2] = rintf( wsz * 0.5f);
      ab.b[a * 4 + 3] = rintf( hsz * 0.5f);
      ++a;
    }
  }

  // 0) one-time layout transforms
  const int xtN = NPP * ICH;
  zero_buf<<<(xtN + 255) / 256, 256, 0, stream>>>(Xt, xtN);
  transpose_feat<<<dim3(157, 8), 256, 0, stream>>>(feat, Xt);
  pack_weights<<<(OCH * ICH * 9 + 255) / 256, 256, 0, stream>>>(rpn_w, Wpack);
  // 1) 3x3 conv (branch-free WMMA GEMM)
  conv3x3_wmma<<<dim3(4, 157), 256, 0, stream>>>(Xt, Wpack, rpn_b, convYt);
  // 2) 1x1 heads (WMMA), stored NHWC-flat as reference's transpose produces
  conv1x1_wmma<<<dim3(3, 157), 32, 0, stream>>>(convYt, reg_w, reg_b, deltas, 36, 0);
  conv1x1_wmma<<<dim3(1, 157), 32, 0, stream>>>(convYt, cls_w, cls_b, scores, 9, 1);
  // 3) exact top-2000 via descending bitonic sort of (score, ~idx) keys
  init_keys<<<NSORT / 256, 256, 0, stream>>>(scores, keys);
  for (int k = 2; k <= NSORT; k <<= 1)
    for (int j = k >> 1; j > 0; j >>= 1)
      bitonic_step<<<NSORT / 256, 256, 0, stream>>>(keys, j, k);
  // 4) decode + clip + min-size filter
  decode_topk<<<(TOPK + 255) / 256, 256, 0, stream>>>(keys, deltas, ab, boxesA, sA, vA);
  // 5) stable valid-first reorder (== argsort(-s) on already-sorted scores)
  reorder_valid_first<<<1, 256, 0, stream>>>(boxesA, sA, vA, boxesB, sB, vB);
  // 6) greedy NMS
  nms_bits<<<(TOPK * NWORDS + 255) / 256, 256, 0, stream>>>(boxesB, mat);
  nms_reduce<<<1, 32, 0, stream>>>(mat, vB, keep);
  // 7) emit (2000,5), zeros where not kept
  finalize<<<(TOPK * 5 + 255) / 256, 256, 0, stream>>>(boxesB, sB, keep, out);
}